// RoiPolling_31602369364098
// MI455X (gfx1250) — compile-verified
//
#include <hip/hip_runtime.h>
#include <hip/hip_bf16.h>

typedef float v2f __attribute__((ext_vector_type(2)));
typedef float v8f __attribute__((ext_vector_type(8)));
typedef float f4  __attribute__((ext_vector_type(4)));
typedef int   v4i __attribute__((ext_vector_type(4)));

#define HAVE_ASYNC (__has_builtin(__builtin_amdgcn_global_load_async_to_lds_b128) && \
                    __has_builtin(__builtin_amdgcn_s_wait_asynccnt))

#if HAVE_ASYNC
typedef __attribute__((address_space(1))) v4i *gv4i_ptr;   // global int4*
typedef __attribute__((address_space(3))) v4i *lv4i_ptr;   // LDS int4*
#endif

namespace {
constexpr int POOL = 7;
constexpr int IH = 200, IW = 200, IC = 512;
constexpr int NROI = 300;
constexpr int NPIX = NROI * POOL * POOL;          // 14700 output pixels
constexpr int TILES = (NPIX + 15) / 16;           // 919 tiles of 16 pixels
}

// One wave handles a tile of 16 output pixels x 16 channels via
// D += diag(w_k) x V_k on v_wmma_f32_16x16x4_f32 (4 corners x 4 K-chunks).
// Block = 128 threads (4 waves -> 4 consecutive 16-channel chunks).
__global__ __launch_bounds__(128)
void roi_bilinear_wmma_kernel(const float* __restrict__ img,
                              const int* __restrict__ rois,
                              float* __restrict__ out) {
  __shared__ unsigned s_addr[4][16][4];   // per-wave: element offset of 4 corners per pixel
  __shared__ float    s_wt[4][16][4];     // per-wave: 4 bilinear weights per pixel
  __shared__ float    s_buf[4][4][16][16];// per-wave: [corner][pixel][channel] tile

  const int lane = threadIdx.x & 31;
  const int wave = threadIdx.x >> 5;
  const int half = lane >> 4;             // 0: lanes 0-15, 1: lanes 16-31
  const int m    = lane & 15;             // pixel row (A/D) == channel col (B/D)

  const int tileBase = blockIdx.x * 16;
  const int cbase    = (blockIdx.y * 4 + wave) * 16;   // this wave's channel chunk

  // ---- per-pixel meta (lanes 0-15): corner offsets + bilinear weights ----
  if (lane < 16) {
    int gp = tileBase + lane;
    if (gp > NPIX - 1) gp = NPIX - 1;     // clamp; stores are guarded later
    const int r   = gp / (POOL * POOL);
    const int rem = gp % (POOL * POOL);
    const int py  = rem / POOL;
    const int px  = rem % POOL;
    const int x0 = rois[r * 4 + 0];
    const int y0 = rois[r * 4 + 1];
    const int w  = rois[r * 4 + 2];
    const int h  = rois[r * 4 + 3];

    const float stepy = (float)h / (float)POOL;
    const float stepx = (float)w / (float)POOL;
    const float ysf = (float)y0 + (float)py * stepy;
    const float xsf = (float)x0 + (float)px * stepx;
    const int ty = (int)floorf(ysf);
    const int tx = (int)floorf(xsf);
    const float fy = ysf - (float)ty;
    const float fx = xsf - (float)tx;
    int by = ty + 1; { const int lim = y0 + h - 1; if (by > lim) by = lim; }
    int bx = tx + 1; { const int lim = x0 + w - 1; if (bx > lim) bx = lim; }

    s_addr[wave][lane][0] = (unsigned)((ty * IW + tx) * IC);
    s_addr[wave][lane][1] = (unsigned)((ty * IW + bx) * IC);
    s_addr[wave][lane][2] = (unsigned)((by * IW + tx) * IC);
    s_addr[wave][lane][3] = (unsigned)((by * IW + bx) * IC);
    s_wt[wave][lane][0] = (1.0f - fy) * (1.0f - fx);
    s_wt[wave][lane][1] = (1.0f - fy) * fx;
    s_wt[wave][lane][2] = fy * (1.0f - fx);
    s_wt[wave][lane][3] = fy * fx;
  }
  // Same-wave LDS ops are in-order (DScnt); no workgroup barrier needed
  // (each wave touches only its own LDS slice).

  // ---- gather corner tiles: 4 corners x 16 pixels x 16 channels ----
  // 64 vectors of 64B => 8 x b128 per lane, contiguous 512B per wavefront store.
#pragma unroll
  for (int i = 0; i < 8; ++i) {
    const int t   = i * 32 + lane;
    const int vec = t >> 2;               // 0..63
    const int sub = t & 3;                // which float4 of the 16-ch vector
    const int k   = vec >> 4;             // corner
    const int p   = vec & 15;             // pixel
    const unsigned off = s_addr[wave][p][k] + (unsigned)(cbase + sub * 4);
    float* dst = &s_buf[wave][k][p][sub * 4];
#if HAVE_ASYNC
    __builtin_amdgcn_global_load_async_to_lds_b128(
        (gv4i_ptr)(v4i*)(img + off), (lv4i_ptr)(v4i*)dst, 0, 0);
#else
    *(f4*)dst = *(const f4*)(img + off);
#endif
  }
#if HAVE_ASYNC
  __builtin_amdgcn_s_wait_asynccnt(0);
#endif

  // ---- combine on the matrix pipe: D += diag(w_k) * V_k ----
  const f4 wt4 = *(const f4*)&s_wt[wave][m][0];
  v8f acc = {};
#pragma unroll
  for (int k = 0; k < 4; ++k) {
    const float wk = wt4[k];
#pragma unroll
    for (int q = 0; q < 4; ++q) {
      // A (16x4 f32): VGPR0 -> K = 0 (lanes 0-15) / 2 (lanes 16-31); VGPR1 -> K = 1 / 3.
      const int ka0 = 2 * half;
      v2f a, b;
      a.x = (m == 4 * q + ka0)     ? wk : 0.0f;
      a.y = (m == 4 * q + ka0 + 1) ? wk : 0.0f;
      // B (4x16 f32): row kk holds V_k[4q+kk][c], c = lane&15.
      b.x = s_buf[wave][k][4 * q + ka0][m];
      b.y = s_buf[wave][k][4 * q + ka0 + 1][m];
      acc = __builtin_amdgcn_wmma_f32_16x16x4_f32(
          /*neg_a=*/false, a, /*neg_b=*/false, b,
          /*c_mod=*/(short)0, acc, /*reuse_a=*/false, /*reuse_b=*/false);
    }
  }

  // ---- store: D VGPR r -> pixel r + 8*half, channel = cbase + (lane&15) ----
  const int c = cbase + m;
#pragma unroll
  for (int r = 0; r < 8; ++r) {
    const int gp = tileBase + r + 8 * half;
    if (gp < NPIX) out[(size_t)gp * IC + c] = acc[r];
  }
}

extern "C" void kernel_launch(void* const* d_in, const int* in_sizes, int n_in,
                              void* d_out, int out_size, void* d_ws, size_t ws_size,
                              hipStream_t stream) {
  const float* img  = (const float*)d_in[0];   // (1,200,200,512) f32
  const int*   rois = (const int*)d_in[1];     // (1,300,4) i32: x,y,w,h
  float*       out  = (float*)d_out;           // (1,300,7,7,512) f32

  dim3 grid(TILES, IC / 64, 1);   // 919 pixel tiles x 8 channel groups
  dim3 block(128, 1, 1);          // 4 waves; each wave = one 16-channel chunk
  roi_bilinear_wmma_kernel<<<grid, block, 0, stream>>>(img, rois, out);
}